// HGNNPConv_11914239279532
// MI455X (gfx1250) — compile-verified
//
#include <hip/hip_runtime.h>
#include <hip/hip_bf16.h>

#define N_VERT   100000
#define N_EDGE   25000
#define NNZ_CNT  3200000
#define IN_CH    256
#define OUT_CH   128

typedef __attribute__((ext_vector_type(2))) float v2f;
typedef __attribute__((ext_vector_type(8))) float v8f;

// ---------------------------------------------------------------------------
// Kernel 1: H = relu(X * W^T + b) using V_WMMA_F32_16X16X4_F32 (exact fp32).
// One wave computes a 32-row x 128-col strip (2 A-frags share each B-frag).
// ---------------------------------------------------------------------------
__global__ __launch_bounds__(256)
void hgnn_gemm_relu(const float* __restrict__ X, const float* __restrict__ W,
                    const float* __restrict__ bias, float* __restrict__ H) {
    const int wave = threadIdx.x >> 5;
    const int lane = threadIdx.x & 31;
    const int pair = blockIdx.x * 8 + wave;      // 32 rows per wave
    const int row0 = pair * 32;
    if (row0 >= N_VERT) return;                   // uniform per wave: EXEC stays all-1s inside

    const int m  = lane & 15;                     // A: row in tile, B/D: col in tile
    const int ks = (lane >> 4) << 1;              // K sub-select {0,2}
    const int mh = (lane >> 4) << 3;              // D: row-half select {0,8}

    const float* xrow0 = X + (size_t)(row0 + m) * IN_CH;
    const float* xrow1 = xrow0 + (size_t)16 * IN_CH;

    v8f z = {};
    v8f acc0[8], acc1[8];
#pragma unroll
    for (int t = 0; t < 8; ++t) { acc0[t] = z; acc1[t] = z; }

    for (int kk = 0; kk < IN_CH; kk += 4) {
        v2f a0 = *(const v2f*)(xrow0 + kk + ks);
        v2f a1 = *(const v2f*)(xrow1 + kk + ks);
#pragma unroll
        for (int t = 0; t < 8; ++t) {
            // B[k][n] = W[t*16 + n][k]   (n == m for this lane)
            v2f bt = *(const v2f*)(W + (size_t)(t * 16 + m) * IN_CH + kk + ks);
            acc0[t] = __builtin_amdgcn_wmma_f32_16x16x4_f32(
                false, a0, false, bt, (short)0, acc0[t], false, false);
            acc1[t] = __builtin_amdgcn_wmma_f32_16x16x4_f32(
                false, a1, false, bt, (short)0, acc1[t], false, false);
        }
    }

#pragma unroll
    for (int t = 0; t < 8; ++t) {
        const float bo = bias[t * 16 + m];
#pragma unroll
        for (int r = 0; r < 8; ++r) {
            float v0 = acc0[t][r] + bo;
            float v1 = acc1[t][r] + bo;
            H[(size_t)(row0 + mh + r) * OUT_CH + t * 16 + m]      = fmaxf(v0, 0.f);
            H[(size_t)(row0 + 16 + mh + r) * OUT_CH + t * 16 + m] = fmaxf(v1, 0.f);
        }
    }
}

// ---------------------------------------------------------------------------
// Kernel 2: degree counts (both directions in one pass).
// ---------------------------------------------------------------------------
__global__ __launch_bounds__(256)
void hgnn_degrees(const int* __restrict__ v_idx, const int* __restrict__ e_idx,
                  float* __restrict__ v_deg, float* __restrict__ e_deg) {
    unsigned i = blockIdx.x * 256u + threadIdx.x;
    if (i >= NNZ_CNT) return;
    atomicAdd(e_deg + e_idx[i], 1.0f);
    atomicAdd(v_deg + v_idx[i], 1.0f);
}

// ---------------------------------------------------------------------------
// Kernel 3: inv[i] = 1 / max(deg[i], 1)
// ---------------------------------------------------------------------------
__global__ __launch_bounds__(256)
void hgnn_invdeg(const float* __restrict__ deg, float* __restrict__ inv, int n) {
    unsigned i = blockIdx.x * 256u + threadIdx.x;
    if (i < (unsigned)n) inv[i] = 1.0f / fmaxf(deg[i], 1.0f);
}

// ---------------------------------------------------------------------------
// Kernel 4: segment scatter-add (one wave per nnz; lane handles 4 channels).
// dst[scatter[i]] += src[gather[i]]   (all 128 channels, float4 per lane)
// ---------------------------------------------------------------------------
__global__ __launch_bounds__(256)
void hgnn_scatter(const float* __restrict__ src, const int* __restrict__ gather_idx,
                  const int* __restrict__ scatter_idx, float* __restrict__ dst) {
    unsigned gid  = blockIdx.x * 256u + threadIdx.x;
    unsigned i    = gid >> 5;
    unsigned lane = gid & 31;
    if (i >= NNZ_CNT) return;
    const int g = gather_idx[i];
    const int s = scatter_idx[i];
    const float4 h = ((const float4*)src)[(size_t)g * 32 + lane];
    float* d = dst + (size_t)s * OUT_CH + lane * 4;
    atomicAdd(d + 0, h.x);
    atomicAdd(d + 1, h.y);
    atomicAdd(d + 2, h.z);
    atomicAdd(d + 3, h.w);
}

// Same, with per-source-row scaling (applies the edge mean on the fly).
__global__ __launch_bounds__(256)
void hgnn_scatter_scaled(const float* __restrict__ src, const float* __restrict__ inv_src,
                         const int* __restrict__ gather_idx, const int* __restrict__ scatter_idx,
                         float* __restrict__ dst) {
    unsigned gid  = blockIdx.x * 256u + threadIdx.x;
    unsigned i    = gid >> 5;
    unsigned lane = gid & 31;
    if (i >= NNZ_CNT) return;
    const int g = gather_idx[i];
    const int s = scatter_idx[i];
    const float sc = inv_src[g];
    float4 h = ((const float4*)src)[(size_t)g * 32 + lane];
    h.x *= sc; h.y *= sc; h.z *= sc; h.w *= sc;
    float* d = dst + (size_t)s * OUT_CH + lane * 4;
    atomicAdd(d + 0, h.x);
    atomicAdd(d + 1, h.y);
    atomicAdd(d + 2, h.z);
    atomicAdd(d + 3, h.w);
}

// ---------------------------------------------------------------------------
// Kernel 5: out = relu(v_sum * inv_v)  (vertex mean + final ReLU)
// ---------------------------------------------------------------------------
__global__ __launch_bounds__(256)
void hgnn_finalize(const float* __restrict__ v_sum, const float* __restrict__ inv_v,
                   float* __restrict__ out) {
    unsigned gid  = blockIdx.x * 256u + threadIdx.x;
    unsigned v    = gid >> 5;
    unsigned lane = gid & 31;
    if (v >= N_VERT) return;
    const float sc = inv_v[v];
    float4 s = ((const float4*)v_sum)[(size_t)v * 32 + lane];
    float4 o;
    o.x = fmaxf(s.x * sc, 0.f);
    o.y = fmaxf(s.y * sc, 0.f);
    o.z = fmaxf(s.z * sc, 0.f);
    o.w = fmaxf(s.w * sc, 0.f);
    ((float4*)out)[(size_t)v * 32 + lane] = o;
}

extern "C" void kernel_launch(void* const* d_in, const int* in_sizes, int n_in,
                              void* d_out, int out_size, void* d_ws, size_t ws_size,
                              hipStream_t stream) {
    const float* X     = (const float*)d_in[0];
    const float* W     = (const float*)d_in[1];
    const float* bias  = (const float*)d_in[2];
    const int*   v_idx = (const int*)d_in[3];
    const int*   e_idx = (const int*)d_in[4];
    float*       out   = (float*)d_out;

    // Workspace layout (floats):
    float* ws    = (float*)d_ws;
    float* H     = ws;                                     // N_VERT*OUT_CH
    float* e_sum = H     + (size_t)N_VERT * OUT_CH;        // N_EDGE*OUT_CH
    float* v_sum = e_sum + (size_t)N_EDGE * OUT_CH;        // N_VERT*OUT_CH
    float* e_deg = v_sum + (size_t)N_VERT * OUT_CH;        // N_EDGE
    float* v_deg = e_deg + N_EDGE;                         // N_VERT
    float* inv_e = v_deg + N_VERT;                         // N_EDGE
    float* inv_v = inv_e + N_EDGE;                         // N_VERT

    // Zero the accumulators (graph-capture-legal async memsets).
    hipMemsetAsync(e_sum, 0, sizeof(float) * (size_t)N_EDGE * OUT_CH, stream);
    hipMemsetAsync(v_sum, 0, sizeof(float) * (size_t)N_VERT * OUT_CH, stream);
    hipMemsetAsync(e_deg, 0, sizeof(float) * (size_t)(N_EDGE + N_VERT), stream);

    // 1) H = relu(X W^T + b): 3125 row-pairs, 8 waves/block.
    hgnn_gemm_relu<<<(3125 + 7) / 8, 256, 0, stream>>>(X, W, bias, H);

    // 2) degrees
    hgnn_degrees<<<(NNZ_CNT + 255) / 256, 256, 0, stream>>>(v_idx, e_idx, v_deg, e_deg);
    hgnn_invdeg<<<(N_EDGE + 255) / 256, 256, 0, stream>>>(e_deg, inv_e, N_EDGE);
    hgnn_invdeg<<<(N_VERT + 255) / 256, 256, 0, stream>>>(v_deg, inv_v, N_VERT);

    // 3) v2e: e_sum[e] += H[v]
    hgnn_scatter<<<(NNZ_CNT * 32u + 255) / 256, 256, 0, stream>>>(H, v_idx, e_idx, e_sum);

    // 4) e2v: v_sum[v] += e_sum[e] * inv_e[e]   (edge mean fused into gather)
    hgnn_scatter_scaled<<<(NNZ_CNT * 32u + 255) / 256, 256, 0, stream>>>(e_sum, inv_e, e_idx, v_idx, v_sum);

    // 5) out = relu(v_sum * inv_v)
    hgnn_finalize<<<(N_VERT * 32u + 255) / 256, 256, 0, stream>>>(v_sum, inv_v, out);
}